// incustomNet_25855703122037
// MI455X (gfx1250) — compile-verified
//
#include <hip/hip_runtime.h>

typedef __attribute__((ext_vector_type(2))) float v2f;
typedef __attribute__((ext_vector_type(8))) float v8f;

#define IN_F  6
#define HID_F 32
#define OUT_F 3

// ---------------- zero workspace accumulators ----------------
__global__ void zero_kernel(float* __restrict__ p, int n) {
    int i = blockIdx.x * blockDim.x + threadIdx.x;
    if (i < n) p[i] = 0.0f;
}

// ---------------- layer-1 edge scatter: msg_sum + degree ----------------
__global__ void edge_agg1(const float* __restrict__ feat,
                          const int* __restrict__ src,
                          const int* __restrict__ dst,
                          float* __restrict__ agg1,
                          float* __restrict__ deg, int E) {
    int e = blockIdx.x * blockDim.x + threadIdx.x;
    if (e >= E) return;
    int s = src[e], d = dst[e];
    const float* f = feat + (size_t)s * IN_F;
    float* a = agg1 + (size_t)d * IN_F;
#pragma unroll
    for (int k = 0; k < IN_F; ++k) atomicAdd(&a[k], f[k]);
    atomicAdd(&deg[d], 1.0f);
}

// ---------------- fused per-node MLP via WMMA ----------------
// Per wave: one 16-node tile.
//   A (16x4 f32): lane = m + 16*(k>=2), vgpr = k%2   (K padded 6 -> 8, two chunks)
//   B (4x16 f32): lane = n + 16*(k>=2), vgpr = k%2   (W1 cols split into two 16-wide tiles)
//   C (16x16 f32): vgpr v, lanes 0-15 -> M=v, lanes 16-31 -> M=v+8, N=lane%16
// h = relu(A@W1 + b1) goes through LDS (layout transpose), then lanes 0-15
// compute z = h @ W2 (no bias; bias added after aggregation).
__global__ __launch_bounds__(64)
void node_mlp(const float* __restrict__ agg1, const float* __restrict__ deg,
              const float* __restrict__ W1,   const float* __restrict__ b1,
              const float* __restrict__ W2,   float* __restrict__ z,
              int n_tiles, int N) {
    __shared__ float lds[2][16][HID_F + 1];

    const int wave = threadIdx.x >> 5;
    const int lane = threadIdx.x & 31;
    const int m  = lane & 15;   // row (A) / col (B,C) index
    const int hi = lane >> 4;   // half-wave -> upper K pair / upper M rows

    int tile = blockIdx.x * 2 + wave;
    const bool tvalid = (tile < n_tiles);
    if (!tvalid) tile = 0;
    int node = tile * 16 + m;
    const bool nvalid = tvalid && (node < N);
    const int rn = node < N ? node : (N - 1);   // clamped for safe loads

    const float dinv = 1.0f / fmaxf(deg[rn], 1.0f);

    // A operands: K chunk 0 = k 0..3, chunk 1 = k 4..7 (6,7 padded 0)
    v2f a0, a1;
    {
        const int k0 = hi * 2;
        a0.x = agg1[(size_t)rn * IN_F + k0]     * dinv;
        a0.y = agg1[(size_t)rn * IN_F + k0 + 1] * dinv;
        const int k4 = 4 + hi * 2;
        a1.x = (k4     < IN_F) ? agg1[(size_t)rn * IN_F + k4]     * dinv : 0.0f;
        a1.y = (k4 + 1 < IN_F) ? agg1[(size_t)rn * IN_F + k4 + 1] * dinv : 0.0f;
    }
    // B operands: W1[k][n], n-tile 0 = cols 0..15, n-tile 1 = cols 16..31
    v2f b00, b01, b10, b11;
    {
        const int n0 = m, n1 = 16 + m;
        const int k0 = hi * 2;
        b00.x = W1[(k0)     * HID_F + n0]; b00.y = W1[(k0 + 1) * HID_F + n0];
        b01.x = W1[(k0)     * HID_F + n1]; b01.y = W1[(k0 + 1) * HID_F + n1];
        const int k4 = 4 + hi * 2;
        b10.x = (k4     < IN_F) ? W1[(k4)     * HID_F + n0] : 0.0f;
        b10.y = (k4 + 1 < IN_F) ? W1[(k4 + 1) * HID_F + n0] : 0.0f;
        b11.x = (k4     < IN_F) ? W1[(k4)     * HID_F + n1] : 0.0f;
        b11.y = (k4 + 1 < IN_F) ? W1[(k4 + 1) * HID_F + n1] : 0.0f;
    }

    v8f c0 = {}, c1 = {};
    c0 = __builtin_amdgcn_wmma_f32_16x16x4_f32(false, a0, false, b00, (short)0, c0, false, false);
    c0 = __builtin_amdgcn_wmma_f32_16x16x4_f32(false, a1, false, b10, (short)0, c0, false, false);
    c1 = __builtin_amdgcn_wmma_f32_16x16x4_f32(false, a0, false, b01, (short)0, c1, false, false);
    c1 = __builtin_amdgcn_wmma_f32_16x16x4_f32(false, a1, false, b11, (short)0, c1, false, false);

    // bias + ReLU, transpose-spill h tile to LDS (row-major [16][32])
    const float bn0 = b1[m];
    const float bn1 = b1[16 + m];
#pragma unroll
    for (int v = 0; v < 8; ++v) {
        const int mm = v + 8 * hi;
        lds[wave][mm][m]      = fmaxf(c0[v] + bn0, 0.0f);
        lds[wave][mm][16 + m] = fmaxf(c1[v] + bn1, 0.0f);
    }
    __syncthreads();

    // z = h @ W2  (3 outputs per node; lanes 0-15 each own one node)
    if (hi == 0 && nvalid) {
        float acc0 = 0.f, acc1 = 0.f, acc2 = 0.f;
#pragma unroll
        for (int j = 0; j < HID_F; ++j) {
            const float hv = lds[wave][m][j];
            acc0 += hv * W2[j * OUT_F + 0];
            acc1 += hv * W2[j * OUT_F + 1];
            acc2 += hv * W2[j * OUT_F + 2];
        }
        z[(size_t)node * OUT_F + 0] = acc0;
        z[(size_t)node * OUT_F + 1] = acc1;
        z[(size_t)node * OUT_F + 2] = acc2;
    }
}

// ---------------- layer-2 edge scatter: only 3 floats per edge ----------------
__global__ void edge_agg2(const float* __restrict__ z,
                          const int* __restrict__ src,
                          const int* __restrict__ dst,
                          float* __restrict__ agg2, int E) {
    int e = blockIdx.x * blockDim.x + threadIdx.x;
    if (e >= E) return;
    int s = src[e], d = dst[e];
    const float* zs = z + (size_t)s * OUT_F;
    float* a = agg2 + (size_t)d * OUT_F;
    atomicAdd(&a[0], zs[0]);
    atomicAdd(&a[1], zs[1]);
    atomicAdd(&a[2], zs[2]);
}

// ---------------- finalize: mean + bias ----------------
__global__ void finalize_kernel(const float* __restrict__ agg2,
                                const float* __restrict__ deg,
                                const float* __restrict__ b2,
                                float* __restrict__ out, int N) {
    int i = blockIdx.x * blockDim.x + threadIdx.x;
    if (i >= N * OUT_F) return;
    int n = i / OUT_F, c = i % OUT_F;
    out[i] = agg2[i] / fmaxf(deg[n], 1.0f) + b2[c];
}

extern "C" void kernel_launch(void* const* d_in, const int* in_sizes, int n_in,
                              void* d_out, int out_size, void* d_ws, size_t ws_size,
                              hipStream_t stream) {
    const float* feat = (const float*)d_in[0];
    const int*   src  = (const int*)d_in[1];
    const int*   dst  = (const int*)d_in[2];
    const float* W1   = (const float*)d_in[3];
    const float* b1   = (const float*)d_in[4];
    const float* W2   = (const float*)d_in[5];
    const float* b2   = (const float*)d_in[6];
    float* out = (float*)d_out;

    const int N = in_sizes[0] / IN_F;
    const int E = in_sizes[1];

    // workspace: [agg1 N*6][deg N][agg2 N*3][z N*3]
    float* agg1 = (float*)d_ws;
    float* deg  = agg1 + (size_t)N * IN_F;
    float* agg2 = deg  + (size_t)N;
    float* zbuf = agg2 + (size_t)N * OUT_F;

    const int zero_n = N * (IN_F + 1 + OUT_F);   // agg1 + deg + agg2 contiguous
    zero_kernel<<<(zero_n + 255) / 256, 256, 0, stream>>>(agg1, zero_n);

    edge_agg1<<<(E + 255) / 256, 256, 0, stream>>>(feat, src, dst, agg1, deg, E);

    const int n_tiles = (N + 15) / 16;
    const int blocks  = (n_tiles + 1) / 2;       // 2 waves (tiles) per 64-thread block
    node_mlp<<<blocks, 64, 0, stream>>>(agg1, deg, W1, b1, W2, zbuf, n_tiles, N);

    edge_agg2<<<(E + 255) / 256, 256, 0, stream>>>(zbuf, src, dst, agg2, E);

    finalize_kernel<<<(N * OUT_F + 255) / 256, 256, 0, stream>>>(agg2, deg, b2, out, N);
}